// PartialConnection_MF_71476845740126
// MI455X (gfx1250) — compile-verified
//
#include <hip/hip_runtime.h>
#include <stdint.h>

// Problem constants (match reference)
#define UNITS  100000
#define N_IN   100000
#define DEGREE 16
#define NEDGE  (UNITS * DEGREE)     // 1,600,000
#define NB     16                   // batch
#define NF     4                    // features
#define BFV    (NB * NF)            // 64 floats per gathered row

typedef __attribute__((ext_vector_type(2))) float v2f;  // native vector for NT store

// ---------------------------------------------------------------------------
// CDNA5 async global<->LDS helpers (CDNA5 ISA, tracked ASYNCcnt)
// ---------------------------------------------------------------------------
__device__ __forceinline__ uint32_t lds_off(const void* p) {
  // Generic shared-aperture address: low 32 bits == LDS byte address (ISA 10.2)
  return (uint32_t)(uintptr_t)p;
}

__device__ __forceinline__ void async_load_b128(uint32_t dst_lds, const void* gaddr) {
  asm volatile("global_load_async_to_lds_b128 %0, %1, off"
               :: "v"(dst_lds), "v"(gaddr) : "memory");
}

__device__ __forceinline__ void async_store_b128(void* gaddr, uint32_t src_lds) {
  asm volatile("global_store_async_from_lds_b128 %0, %1, off"
               :: "v"(gaddr), "v"(src_lds) : "memory");
}

__device__ __forceinline__ void wait_async0() {
  asm volatile("s_wait_asynccnt 0" ::: "memory");
}

// ---------------------------------------------------------------------------
// Kernel 1: transpose x (B, N_IN, F) -> xt (N_IN, B*F) as a pure async DMA
// pipeline: global -> LDS tile (async load) -> global (async store). No VGPR
// data movement; VALU only computes addresses.
// ---------------------------------------------------------------------------
#define NT 128   // n-rows per block

__global__ __launch_bounds__(256) void xpose_kernel(const float* __restrict__ x,
                                                    float* __restrict__ xt) {
  __shared__ __align__(16) float4 tile[NT * 17];  // pad 16->17 vs bank conflicts
  const int tid = threadIdx.x;
  const int n0  = blockIdx.x * NT;

  // Load phase: 8 passes, 2 batches per pass; global reads are contiguous
  // 2KB runs per half-block, DMA'd straight into LDS.
#pragma unroll
  for (int j = 0; j < 8; ++j) {
    const int b  = j * 2 + (tid >> 7);
    const int nl = tid & (NT - 1);
    const int n  = n0 + nl;
    if (n < N_IN) {
      async_load_b128(lds_off(&tile[nl * 17 + b]),
                      x + (size_t)b * (N_IN * NF) + (size_t)n * NF);
    }
  }
  wait_async0();       // this wave's async LDS writes complete
  __syncthreads();     // all waves' LDS writes visible

  // Store phase: each pass DMAs a contiguous 4KB run of xt from LDS.
#pragma unroll
  for (int j = 0; j < 8; ++j) {
    const int q  = j * 256 + tid;
    const int nl = q >> 4;
    const int b  = q & 15;
    const int n  = n0 + nl;
    if (n < N_IN) {
      async_store_b128(xt + (size_t)n * BFV + b * NF,
                       lds_off(&tile[nl * 17 + b]));
    }
  }
  // S_ENDPGM performs an implicit wait-idle, draining ASYNCcnt before the
  // dependent kernel launches on the stream.
}

// ---------------------------------------------------------------------------
// Kernel 2: one wave per unit. Stage 64 units' edge metadata (idx/k/bias,
// 12KB) into LDS with async b128 copies, then each wave reduces its unit's
// 16 edges: lane l accumulates output elements (2l, 2l+1) of the 64-float
// (b,f) vector. Gather reads are 256B coalesced per wave from L2-resident xt.
// Output stores are non-temporal to keep xt resident in L2.
// ---------------------------------------------------------------------------
#define UB 64              // units per block
#define EB (UB * DEGREE)   // 1024 edges per block

__global__ __launch_bounds__(256) void gather_seg_kernel(const float* __restrict__ xt,
                                                         const float* __restrict__ kern,
                                                         const float* __restrict__ bias,
                                                         const int*   __restrict__ src_idx,
                                                         float* __restrict__ out) {
  __shared__ __align__(16) int   s_idx[EB];
  __shared__ __align__(16) float s_k[EB];
  __shared__ __align__(16) float s_b[EB];

  const int  tid = threadIdx.x;
  const int  u0  = blockIdx.x * UB;
  const long e0  = (long)u0 * DEGREE;

  {
    const long e = e0 + (long)tid * 4;
    if (e + 3 < NEDGE) {   // contiguous 16B per lane per array
      async_load_b128(lds_off(&s_idx[tid * 4]), src_idx + e);
      async_load_b128(lds_off(&s_k[tid * 4]),   kern + e);
      async_load_b128(lds_off(&s_b[tid * 4]),   bias + e);
    }
  }
  wait_async0();       // this wave's async copies done
  __syncthreads();     // all waves' copies visible

  const int wave = tid >> 5;
  const int lane = tid & 31;
  const int c    = lane * 2;      // element index 0..62 of the 64-float row
  const int b    = c >> 2;        // batch
  const int f    = c & 3;         // feature (0 or 2; v2f covers f, f+1)

#pragma unroll 1
  for (int i = 0; i < 8; ++i) {
    const int ul = wave + i * 8;  // adjacent waves -> adjacent units (store locality)
    const int u  = u0 + ul;
    if (u >= UNITS) continue;

    float a0 = 0.0f, a1 = 0.0f, bs = 0.0f;
    const int eb = ul * DEGREE;

#pragma unroll
    for (int d = 0; d < DEGREE; ++d) {
      const int   idx = s_idx[eb + d];   // wave-uniform LDS broadcast
      const float k   = s_k[eb + d];
      bs += s_b[eb + d];
      const v2f g = *(const v2f*)(xt + (size_t)idx * BFV + c);
      a0 = fmaf(g.x, k, a0);
      a1 = fmaf(g.y, k, a1);
    }

    v2f r;
    r.x = a0 + bs;
    r.y = a1 + bs;
    __builtin_nontemporal_store(
        r, (v2f*)(out + (size_t)b * (UNITS * NF) + (size_t)u * NF + f));
  }
}

// ---------------------------------------------------------------------------
// Fallback (if workspace can't hold transposed x): direct gather, one thread
// per (b, u), float4 accumulation, contiguous float4 stores.
// ---------------------------------------------------------------------------
__global__ __launch_bounds__(256) void direct_kernel(const float* __restrict__ x,
                                                     const float* __restrict__ kern,
                                                     const float* __restrict__ bias,
                                                     const int*   __restrict__ src_idx,
                                                     float* __restrict__ out) {
  const long t = (long)blockIdx.x * blockDim.x + threadIdx.x;
  if (t >= (long)NB * UNITS) return;
  const int b = (int)(t / UNITS);
  const int u = (int)(t % UNITS);

  const float* xb = x + (size_t)b * N_IN * NF;
  const int e = u * DEGREE;
  float4 acc = make_float4(0.f, 0.f, 0.f, 0.f);
  float  bs  = 0.f;
#pragma unroll
  for (int d = 0; d < DEGREE; ++d) {
    const int   idx = src_idx[e + d];
    const float k   = kern[e + d];
    bs += bias[e + d];
    const float4 g = *(const float4*)(xb + (size_t)idx * NF);
    acc.x = fmaf(g.x, k, acc.x);
    acc.y = fmaf(g.y, k, acc.y);
    acc.z = fmaf(g.z, k, acc.z);
    acc.w = fmaf(g.w, k, acc.w);
  }
  acc.x += bs; acc.y += bs; acc.z += bs; acc.w += bs;
  *(float4*)(out + (size_t)b * UNITS * NF + (size_t)u * NF) = acc;
}

// ---------------------------------------------------------------------------
extern "C" void kernel_launch(void* const* d_in, const int* in_sizes, int n_in,
                              void* d_out, int out_size, void* d_ws, size_t ws_size,
                              hipStream_t stream) {
  const float* x       = (const float*)d_in[0];
  const float* kern    = (const float*)d_in[1];
  const float* bias    = (const float*)d_in[2];
  const int*   src_idx = (const int*)d_in[3];
  // d_in[4] = seg_ids: implicit (u = e / DEGREE), not needed.
  float* out = (float*)d_out;

  const size_t xt_bytes = (size_t)N_IN * BFV * sizeof(float);  // 25.6 MB
  if (d_ws != nullptr && ws_size >= xt_bytes) {
    float* xt = (float*)d_ws;
    xpose_kernel<<<(N_IN + NT - 1) / NT, 256, 0, stream>>>(x, xt);
    gather_seg_kernel<<<(UNITS + UB - 1) / UB, 256, 0, stream>>>(xt, kern, bias, src_idx, out);
  } else {
    const long tot = (long)NB * UNITS;
    direct_kernel<<<(int)((tot + 255) / 256), 256, 0, stream>>>(x, kern, bias, src_idx, out);
  }
}